// DNAConvEncoder_62053687493152
// MI455X (gfx1250) — compile-verified
//
#include <hip/hip_runtime.h>
#include <hip/hip_bf16.h>

// ---------------------------------------------------------------------------
// TAGConv x3 GNN for MI455X (gfx1250, wave32).
//  - SpMM hops: 1 wave per edge, lanes cover the feature row, f32 global atomics.
//  - Dense per-hop GEMMs: native fp32 WMMA (v_wmma_f32_16x16x4_f32), one wave
//    computes one 16x16 output tile; 16 x IND weight slice DMA'd into LDS by
//    the Tensor Data Mover (tensor_load_to_lds + s_wait_tensorcnt).
//  - Bias / accumulation fused into the WMMA C operand.
// ---------------------------------------------------------------------------

typedef __attribute__((ext_vector_type(2))) float    v2f;
typedef __attribute__((ext_vector_type(8))) float    v8f;
typedef __attribute__((ext_vector_type(4))) unsigned v4u;
typedef __attribute__((ext_vector_type(4))) int      v4i;
typedef __attribute__((ext_vector_type(8))) int      v8i;

// ---------------------------- utility kernels ------------------------------

__global__ void zero_kernel(float* __restrict__ p, long n) {
    long i = (long)blockIdx.x * blockDim.x + threadIdx.x;
    long stride = (long)gridDim.x * blockDim.x;
    for (; i < n; i += stride) p[i] = 0.0f;
}

__global__ void deg_kernel(const int* __restrict__ col, float* __restrict__ deg, int E) {
    int e = blockIdx.x * blockDim.x + threadIdx.x;
    if (e < E) atomicAdd(&deg[col[e]], 1.0f);
}

__global__ void dinv_kernel(float* __restrict__ deg, int N) {
    int i = blockIdx.x * blockDim.x + threadIdx.x;
    if (i < N) {
        float d = deg[i];
        deg[i] = (d > 0.0f) ? rsqrtf(d) : 0.0f;
    }
}

__global__ void norm_kernel(const int* __restrict__ row, const int* __restrict__ col,
                            const float* __restrict__ dinv, float* __restrict__ nrm, int E) {
    int e = blockIdx.x * blockDim.x + threadIdx.x;
    if (e < E) nrm[e] = dinv[row[e]] * dinv[col[e]];
}

// ------------------------------- SpMM hop ----------------------------------
// h_out[col[e], :] += norm[e] * h_in[row[e], :]   (one wave32 per edge)

template <int IND>
__global__ void spmm_kernel(const int* __restrict__ row, const int* __restrict__ col,
                            const float* __restrict__ nrm,
                            const float* __restrict__ Hin, float* __restrict__ Hout, int E) {
    int wave = (int)(((long)blockIdx.x * blockDim.x + threadIdx.x) >> 5);
    int lane = threadIdx.x & 31;
    if (wave >= E) return;
    int r = row[wave];
    int c = col[wave];
    float w = nrm[wave];
    const float* __restrict__ src = Hin + (long)r * IND;
    float* __restrict__ dst = Hout + (long)c * IND;
#pragma unroll
    for (int j = 0; j < IND / 32; ++j) {
        int f = lane + j * 32;                 // lane-contiguous -> coalesced
        atomicAdd(&dst[f], w * src[f]);        // global_atomic_add_f32
    }
}

// ----------------------- dense hop GEMM via f32 WMMA -----------------------
// Out[n, o] (+)= sum_d H[n, d] * Wk[o, d]   ( == H @ Wk^T, einsum 'nd,od->no')
// One wave -> one 16(M=nodes) x 16(N=out feats) tile, K streamed 4 at a time.
// Weight slice staged in LDS by the Tensor Data Mover (wave 0 programs the D#).

template <int IND>
__global__ __launch_bounds__(256) void tag_gemm_wmma(
        const float* __restrict__ H, const float* __restrict__ Wk,
        const float* __restrict__ bias, float* __restrict__ Out,
        int outD, int nTiles, int accum) {
    __shared__ float Ws[16 * IND];             // weight slice: rows o0..o0+15
    const int o0 = blockIdx.y * 16;

    // --- TDM: DMA a 2D tile (16 rows x IND cols, 4B elems) of Wk into LDS ---
    if (threadIdx.x < 32) {
        unsigned lds = (unsigned)(unsigned long long)(size_t)(const void*)Ws;  // flat low 32b = LDS offset
        unsigned long long ga = (unsigned long long)(size_t)(const void*)(Wk + (long)o0 * IND);
        v4u g0;
        g0[0] = 1u;                                            // count=1, user mode, no gather
        g0[1] = lds;                                           // lds_addr
        g0[2] = (unsigned)(ga & 0xFFFFFFFFu);                  // global_addr[31:0]
        g0[3] = (unsigned)((ga >> 32) & 0x1FFFFFFu) | (2u << 30); // global_addr[56:32], type=2
        v8i g1;
        g1[0] = (int)(2u << 16);                               // data_size=2 (4B); no multicast/pad
        g1[1] = (int)((unsigned)(IND & 0xFFFF) << 16);         // tensor_dim0[15:0] = IND
        g1[2] = (int)(16u << 16);                              // tensor_dim0 hi=0; tensor_dim1[15:0]=16
        g1[3] = (int)((unsigned)IND << 16);                    // tensor_dim1 hi=0; tile_dim0=IND
        g1[4] = 16;                                            // tile_dim1=16; tile_dim2=0 (2D)
        g1[5] = IND;                                           // tensor_dim0_stride[31:0] = IND
        g1[6] = 0;                                             // stride hi / dim1_stride lo
        g1[7] = 0;                                             // dim1_stride hi (unused, 2D)
        v4i gz = {0, 0, 0, 0};
#if __clang_major__ >= 23
        v8i gz8 = {0, 0, 0, 0, 0, 0, 0, 0};
        __builtin_amdgcn_tensor_load_to_lds(g0, g1, gz, gz, gz8, 0);
#else
        __builtin_amdgcn_tensor_load_to_lds(g0, g1, gz, gz, 0);
#endif
        __builtin_amdgcn_s_wait_tensorcnt(0);                  // s_wait_tensorcnt 0
    }
    __syncthreads();                                           // release other waves

    const int wave  = threadIdx.x >> 5;
    const int tileM = blockIdx.x * 8 + wave;   // 256 threads = 8 waves/block
    if (tileM >= nTiles) return;               // wave-uniform guard (EXEC all-1s below)

    const int lane = threadIdx.x & 31;
    const int half = lane >> 4;                // 0: K=d0,d0+1   1: K=d0+2,d0+3
    const int l16  = lane & 15;
    const int oc   = o0 + l16;

    // C/D layout (ISA 7.12.2): VGPR g, lanes<16 -> M=g, lanes>=16 -> M=g+8.
    v8f acc;
    if (accum) {
#pragma unroll
        for (int g = 0; g < 8; ++g)
            acc[g] = Out[(long)(tileM * 16 + g + 8 * half) * outD + oc];
    } else {
        float bv = bias[oc];
#pragma unroll
        for (int g = 0; g < 8; ++g) acc[g] = bv;
    }

    const float* __restrict__ hrow = H  + (long)(tileM * 16 + l16) * IND;
    const float* __restrict__ wrow = Ws + l16 * IND;

#pragma unroll 8
    for (int d0 = 0; d0 < IND; d0 += 4) {
        const int ka = d0 + 2 * half;
        v2f A; A[0] = hrow[ka]; A[1] = hrow[ka + 1];   // A 16x4 f32 layout
        v2f B; B[0] = wrow[ka]; B[1] = wrow[ka + 1];   // B 4x16 f32 layout
        acc = __builtin_amdgcn_wmma_f32_16x16x4_f32(
            /*neg_a=*/false, A, /*neg_b=*/false, B,
            /*c_mod=*/(short)0, acc, /*reuse_a=*/false, /*reuse_b=*/false);
    }

#pragma unroll
    for (int g = 0; g < 8; ++g)
        Out[(long)(tileM * 16 + g + 8 * half) * outD + oc] = acc[g];
}

// ------------------------------ epilogues ----------------------------------
// Deterministic hash RNG stands in for JAX threefry (not reproducible on-device).

__device__ __forceinline__ unsigned hash_u32(unsigned x) {
    x ^= x >> 16; x *= 0x7feb352du;
    x ^= x >> 15; x *= 0x846ca68bu;
    x ^= x >> 16;
    return x;
}

__global__ void dropout_kernel(const float* __restrict__ src, float* __restrict__ dst,
                               long n, unsigned seed) {
    long i = (long)blockIdx.x * blockDim.x + threadIdx.x;
    long stride = (long)gridDim.x * blockDim.x;
    for (; i < n; i += stride) {
        unsigned h = hash_u32((unsigned)i ^ seed);
        dst[i] = (h & 1u) ? src[i] * 2.0f : 0.0f;   // p = 0.5, scale 1/(1-p)
    }
}

__global__ void elu_dropout_kernel(const float* __restrict__ src, float* __restrict__ dst,
                                   long n, unsigned seed) {
    long i = (long)blockIdx.x * blockDim.x + threadIdx.x;
    long stride = (long)gridDim.x * blockDim.x;
    for (; i < n; i += stride) {
        float x = src[i];
        x = (x > 0.0f) ? x : (expf(x) - 1.0f);      // ELU
        unsigned h = hash_u32((unsigned)i ^ seed);
        dst[i] = (h & 1u) ? x * 2.0f : 0.0f;
    }
}

// ------------------------------- launcher ----------------------------------

extern "C" void kernel_launch(void* const* d_in, const int* in_sizes, int n_in,
                              void* d_out, int out_size, void* d_ws, size_t ws_size,
                              hipStream_t stream) {
    const float* x  = (const float*)d_in[0];
    const int*   ei = (const int*)d_in[1];     // jnp.int64 -> int32 under default JAX x32
    const float* W1 = (const float*)d_in[2];
    const float* b1 = (const float*)d_in[3];
    const float* W2 = (const float*)d_in[4];
    const float* b2 = (const float*)d_in[5];
    const float* W3 = (const float*)d_in[6];
    const float* b3 = (const float*)d_in[7];

    const int N = in_sizes[0] / 128;           // 100000 (divisible by 16)
    const int E = in_sizes[1] / 2;             // 1600000
    const int* row = ei;
    const int* col = ei + E;

    float* ws   = (float*)d_ws;
    float* dinv = ws;                          // N   (degree, then 1/sqrt(deg))
    float* nrm  = dinv + N;                    // E
    float* bufA = nrm + E;                     // N*128
    float* bufB = bufA + (long)N * 128;        // N*128
    float* bufO = bufB + (long)N * 128;        // N*128 (GEMM accumulator)

    const int nTiles = N / 16;
    const dim3 blk(256);

    // gcn_norm (no self loops): deg, 1/sqrt, edge weights
    zero_kernel<<<1024, blk, 0, stream>>>(dinv, N);
    deg_kernel<<<(E + 255) / 256, blk, 0, stream>>>(col, dinv, E);
    dinv_kernel<<<(N + 255) / 256, blk, 0, stream>>>(dinv, N);
    norm_kernel<<<(E + 255) / 256, blk, 0, stream>>>(row, col, dinv, nrm, E);

    const int spmmBlocks = (int)(((long)E * 32 + 255) / 256);

    // -------- layer 1: 128 -> 128, K=3, then dropout --------
    {
        dim3 grid((nTiles + 7) / 8, 128 / 16);
        tag_gemm_wmma<128><<<grid, blk, 0, stream>>>(x, W1, b1, bufO, 128, nTiles, 0);
        const float* cur = x;
        float* nxt = bufA;
        for (int k = 1; k <= 3; ++k) {
            zero_kernel<<<4096, blk, 0, stream>>>(nxt, (long)N * 128);
            spmm_kernel<128><<<spmmBlocks, blk, 0, stream>>>(row, col, nrm, cur, nxt, E);
            tag_gemm_wmma<128><<<grid, blk, 0, stream>>>(nxt, W1 + (long)k * 128 * 128,
                                                         nullptr, bufO, 128, nTiles, 1);
            cur = nxt;
            nxt = (nxt == bufA) ? bufB : bufA;
        }
        dropout_kernel<<<4096, blk, 0, stream>>>(bufO, bufB, (long)N * 128, 0x9E3779B9u);
    }

    // -------- layer 2: 128 -> 64, K=3, then ELU + dropout --------
    {
        dim3 grid((nTiles + 7) / 8, 64 / 16);
        tag_gemm_wmma<128><<<grid, blk, 0, stream>>>(bufB, W2, b2, bufO, 64, nTiles, 0);
        const float* cur = bufB;
        float* nxt = bufA;
        for (int k = 1; k <= 3; ++k) {
            zero_kernel<<<4096, blk, 0, stream>>>(nxt, (long)N * 128);
            spmm_kernel<128><<<spmmBlocks, blk, 0, stream>>>(row, col, nrm, cur, nxt, E);
            tag_gemm_wmma<128><<<grid, blk, 0, stream>>>(nxt, W2 + (long)k * 64 * 128,
                                                         nullptr, bufO, 64, nTiles, 1);
            cur = nxt;
            nxt = (nxt == bufA) ? bufB : bufA;
        }
        elu_dropout_kernel<<<4096, blk, 0, stream>>>(bufO, bufB, (long)N * 64, 0x85EBCA6Bu);
    }

    // -------- layer 3: 64 -> 16, K=3, accumulate straight into d_out --------
    {
        float* out = (float*)d_out;
        dim3 grid((nTiles + 7) / 8, 16 / 16);
        tag_gemm_wmma<64><<<grid, blk, 0, stream>>>(bufB, W3, b3, out, 16, nTiles, 0);
        const float* cur = bufB;
        float* nxt = bufA;
        for (int k = 1; k <= 3; ++k) {
            zero_kernel<<<4096, blk, 0, stream>>>(nxt, (long)N * 64);
            spmm_kernel<64><<<spmmBlocks, blk, 0, stream>>>(row, col, nrm, cur, nxt, E);
            tag_gemm_wmma<64><<<grid, blk, 0, stream>>>(nxt, W3 + (long)k * 16 * 64,
                                                        nullptr, out, 16, nTiles, 1);
            cur = nxt;
            nxt = (nxt == bufA) ? bufB : bufA;
        }
    }
}